// BinBCELoss_6648609374567
// MI455X (gfx1250) — compile-verified
//
#include <hip/hip_runtime.h>
#include <hip/hip_bf16.h>

typedef __attribute__((ext_vector_type(2))) float v2f;
typedef __attribute__((ext_vector_type(8))) float v8f;

// ---------------------------------------------------------------------------
// Exact full-wave32 sum using the f32 WMMA unit.
// A (16x4):  lane m(<16): A[m,0]=acc_m, A[m,1]=0 ; lane 16+m: A[m,2]=acc_{16+m}
// B (4x16):  all ones  =>  D[m,n] = acc_m + acc_{16+m}  (rowsum, any column)
// Lane l<16 holds D rows 0..7 in its 8 VGPRs, lane l>=16 holds rows 8..15.
// sum(own 8 VGPRs) + shfl_xor(16) == sum over all 32 lanes, exact in fp32.
// Requires EXEC == all ones (call only from fully converged code).
// ---------------------------------------------------------------------------
__device__ __forceinline__ float wave_reduce_wmma(float acc) {
    v2f a; a.x = acc;  a.y = 0.0f;
    v2f b; b.x = 1.0f; b.y = 1.0f;
    v8f c = {};
    v8f d = __builtin_amdgcn_wmma_f32_16x16x4_f32(
        /*neg_a=*/false, a, /*neg_b=*/false, b,
        /*c_mod=*/(short)0, c, /*reuse_a=*/false, /*reuse_b=*/false);
    float s = ((d[0] + d[1]) + (d[2] + d[3])) + ((d[4] + d[5]) + (d[6] + d[7]));
    s += __shfl_xor(s, 16, 32);
    return s;
}

// softplus(x) - x * (j < tgt), numerically stable:
// softplus(x) = max(x,0) + log(1 + exp(-|x|))
__device__ __forceinline__ float sp_term(float x, int j, int tgt) {
    float e  = __expf(-fabsf(x));          // v_exp_f32 (+ free src modifiers)
    float sp = fmaxf(x, 0.0f) + __logf(1.0f + e);  // v_log_f32
    return sp - ((j < tgt) ? x : 0.0f);    // v_cndmask + v_sub
}

// One block (256 threads = 8 waves) per row. 128-bit streaming loads.
__global__ __launch_bounds__(256)
void bce_row_kernel(const float* __restrict__ in,
                    const int*   __restrict__ targets,
                    float*       __restrict__ partials,
                    int N) {
    const int row  = blockIdx.x;
    const int tid  = threadIdx.x;
    const int tgt  = targets[row];                  // one scalar per row (L2 hit)
    const float* rowp = in + (size_t)row * (size_t)N;

    float local = 0.0f;

    if ((N & 3) == 0) {
        const int N4 = N >> 2;
        const float4* rp = reinterpret_cast<const float4*>(rowp);
        for (int c = tid; c < N4; c += 256) {       // global_load_b128 stream
            float4 v = rp[c];
            int j = c << 2;
            local += sp_term(v.x, j,     tgt);
            local += sp_term(v.y, j + 1, tgt);
            local += sp_term(v.z, j + 2, tgt);
            local += sp_term(v.w, j + 3, tgt);
        }
    } else {
        for (int j = tid; j < N; j += 256)
            local += sp_term(rowp[j], j, tgt);
    }

    // fully converged here -> WMMA wave reduction is legal (EXEC all ones)
    float wsum = wave_reduce_wmma(local);

    __shared__ float lred[8];
    const int wave = tid >> 5;
    const int lane = tid & 31;
    if (lane == 0) lred[wave] = wsum;
    __syncthreads();
    if (tid == 0) {
        float t = ((lred[0] + lred[1]) + (lred[2] + lred[3]))
                + ((lred[4] + lred[5]) + (lred[6] + lred[7]));
        partials[row] = t;
    }
}

// Single-block finalize: sum B partials in double, scale, write f32 scalar.
__global__ __launch_bounds__(256)
void bce_finalize_kernel(const float* __restrict__ partials,
                         int nPart, float inv_n,
                         float* __restrict__ out) {
    const int tid = threadIdx.x;
    double local = 0.0;
    for (int i = tid; i < nPart; i += 256)
        local += (double)partials[i];

    __shared__ double red[256];
    red[tid] = local;
    __syncthreads();
    if (tid == 0) {
        double s = 0.0;
        for (int i = 0; i < 256; ++i) s += red[i];
        out[0] = (float)(s * (double)inv_n);
    }
}

extern "C" void kernel_launch(void* const* d_in, const int* in_sizes, int n_in,
                              void* d_out, int out_size, void* d_ws, size_t ws_size,
                              hipStream_t stream) {
    const float* in      = (const float*)d_in[0];
    const int*   targets = (const int*)d_in[1];   // harness: integer -> const int*
    const int total = in_sizes[0];                // B*N = 67,108,864 (< 2^31)
    const int B     = in_sizes[1];                // 8192 rows
    const int N     = total / B;                  // 8192 cols

    float* partials = (float*)d_ws;               // B floats of scratch

    bce_row_kernel<<<dim3(B), dim3(256), 0, stream>>>(in, targets, partials, N);

    const float inv_n = (float)(1.0 / (double)total);
    bce_finalize_kernel<<<dim3(1), dim3(256), 0, stream>>>(partials, B, inv_n,
                                                           (float*)d_out);
}